// EdgePruner_19954418057837
// MI455X (gfx1250) — compile-verified
//
#include <hip/hip_runtime.h>
#include <hip/hip_bf16.h>
#include <stdint.h>

typedef __attribute__((ext_vector_type(2))) float v2f;
typedef __attribute__((ext_vector_type(8))) float v8f;
typedef __attribute__((ext_vector_type(4))) unsigned int u32x4;
typedef __attribute__((ext_vector_type(8))) int i32x8;
typedef __attribute__((ext_vector_type(4))) int i32x4;

#define NB 16
#define NP 512
#define NM 13          // L+1
#define ND 768
#define NLH 144        // L*H
#define NDH 64
#define NK 9216        // L*H*DH

#define BLK_ROWS 64
#define THREADS 512    // 16 waves
#define NCT 48         // 768/16 col tiles
#define CT_PER_WAVE 3
#define RT_PER_WAVE 4
#define LDSA_STRIDE 66 // 64 + 2 pad (TDM: pad_interval=64dw, pad_amount=2dw)

// packed-B strides (floats)
#define B_KK_STRIDE (NCT * 64)          // 3072 floats per k-step
#define B_CHUNK_STRIDE (16 * B_KK_STRIDE) // 49152 floats per 64-k chunk

#if defined(__has_builtin)
#if __has_builtin(__builtin_amdgcn_tensor_load_to_lds) && \
    __has_builtin(__builtin_amdgcn_s_wait_tensorcnt)
#define USE_TDM 1
#endif
#endif

// ---------------------------------------------------------------------------
// Kernel 1: per-batch constant term
// ---------------------------------------------------------------------------
__global__ __launch_bounds__(256) void base_kernel(
    const float* __restrict__ mlp_mask,    // [B, M]
    const float* __restrict__ attn_mask,   // [B, L*H]
    const float* __restrict__ mlp_const,   // [M, D]
    const float* __restrict__ attn_const,  // [L*H, D]
    const float* __restrict__ post_bias,   // [D]
    float* __restrict__ base)              // [B, D]
{
    int idx = blockIdx.x * blockDim.x + threadIdx.x;
    if (idx >= NB * ND) return;
    int b = idx / ND;
    int d = idx - b * ND;
    float s = post_bias[d];
    #pragma unroll
    for (int m = 0; m < NM; ++m)
        s += (1.0f - mlp_mask[b * NM + m]) * mlp_const[m * ND + d];
    for (int lh = 0; lh < NLH; ++lh)
        s += (1.0f - attn_mask[b * NLH + lh]) * attn_const[lh * ND + d];
    base[idx] = s;
}

// ---------------------------------------------------------------------------
// Kernel 2: pack W_O [9216, 768] into WMMA B-fragment lane order so the hot
// loop loads one coalesced b64 per fragment at a constant immediate offset.
//   fragment (kstep s = k>>2, col tile ct): lane = (d&15) + ((k&2)<<3),
//   reg = k&1 ; packed[((s*48+ct)*32 + lane)*2 + reg]
// ---------------------------------------------------------------------------
__global__ __launch_bounds__(256) void pack_kernel(
    const float* __restrict__ W_O, float* __restrict__ packed)
{
    int idx = blockIdx.x * blockDim.x + threadIdx.x;
    if (idx >= NK * ND) return;
    int k = idx / ND;
    int d = idx - k * ND;
    int s    = k >> 2;
    int koff = k & 3;
    int lane = (d & 15) + ((koff & 2) << 3);
    int reg  = koff & 1;
    int ct   = d >> 4;
    packed[(((size_t)s * NCT + ct) * 32 + lane) * 2 + reg] = W_O[idx];
}

// ---------------------------------------------------------------------------
// Kernel 3: fused  out = attn GEMM (WMMA f32 16x16x4) + masked mlp sum + base
//   Block: 64 rows x 768 cols, 16 waves; wave = 4 row-tiles x 3 col-tiles.
//   A chunk (64 rows x 64 k) DMA'd to LDS by the Tensor Data Mover, double
//   buffered; attn_mask folded into the (much smaller) B fragment.
// ---------------------------------------------------------------------------
__global__ __launch_bounds__(THREADS) void fused_kernel(
    const float* __restrict__ mlp_cache,   // [B,P,M,D]
    const float* __restrict__ attn_cache,  // [row, 9216]
    const float* __restrict__ mlp_mask,    // [B, M]
    const float* __restrict__ attn_mask,   // [B, L*H]
    const float* __restrict__ packedB,     // repacked W_O
    const float* __restrict__ base,        // [B, D]
    float* __restrict__ out)               // [B*P, D]
{
    __shared__ float ldsA[2][BLK_ROWS * LDSA_STRIDE];

    const int row0  = blockIdx.x * BLK_ROWS;
    const int b     = row0 >> 9;               // P=512 rows per batch
    const int tid   = threadIdx.x;
    const int wave  = tid >> 5;
    const int lane  = tid & 31;
    const int lhalf = lane >> 4;
    const int l16   = lane & 15;

    v8f acc[RT_PER_WAVE][CT_PER_WAVE];
    #pragma unroll
    for (int rt = 0; rt < RT_PER_WAVE; ++rt)
        #pragma unroll
        for (int t = 0; t < CT_PER_WAVE; ++t)
            acc[rt][t] = (v8f){0.f,0.f,0.f,0.f,0.f,0.f,0.f,0.f};

    const float* amask = attn_mask + b * NLH;
    // loop-carried per-lane B base: fragment (kk,t) sits at constant offset
    // kk*3072 + t*64 floats from here; advance by 49152 floats per chunk.
    const float* bb = packedB + ((size_t)(wave * CT_PER_WAVE) * 32 + lane) * 2;

#ifdef USE_TDM
    const uint64_t abase =
        (uint64_t)(uintptr_t)attn_cache + (uint64_t)row0 * NK * 4u;
    auto tdm_issue = [&](int c) {
        // D# group0: count=1 | lds_addr | global_addr[56:0] | type=2
        uint64_t ga  = abase + (uint64_t)c * NDH * 4u;
        uint32_t lds = (uint32_t)(size_t)(&ldsA[c & 1][0]);
        u32x4 g0 = { 1u, lds, (uint32_t)ga,
                     (uint32_t)(ga >> 32) | (2u << 30) };
        // D# group1: data_size=4B, pad_enable, pad_interval=64dw, pad_amount=2dw
        //   tensor_dim0=9216, tensor_dim1=8192, tile_dim0=64, tile_dim1=64,
        //   tensor_dim0_stride=9216
        i32x8 g1 = { (int)((2u << 16) | (1u << 20) | (5u << 22) | (1u << 25)),
                     (int)((NK & 0xFFFF) << 16),
                     (int)((NK >> 16) | ((8192 & 0xFFFF) << 16)),
                     (int)(NDH << 16),
                     (int)NDH,
                     (int)NK,
                     0, 0 };
        i32x4 gz4 = { 0, 0, 0, 0 };
        i32x8 gz8 = { 0, 0, 0, 0, 0, 0, 0, 0 };
        __builtin_amdgcn_tensor_load_to_lds(g0, g1, gz4, gz4, gz8, 0);
    };
    if (wave == 0) tdm_issue(0);
#endif

    #pragma unroll 1
    for (int c = 0; c < NLH; ++c) {
#ifdef USE_TDM
        if (wave == 0) {
            if (c + 1 < NLH) {
                tdm_issue(c + 1);                      // prefetch next chunk
                __builtin_amdgcn_s_wait_tensorcnt(1);  // chunk c complete
            } else {
                __builtin_amdgcn_s_wait_tensorcnt(0);
            }
        }
#else
        {   // cooperative staging fallback: 512 threads x 8 floats
            const int st_row = tid >> 3;
            const int st_seg = (tid & 7) * 8;
            const float* src =
                attn_cache + ((size_t)(row0 + st_row) * NLH + c) * NDH + st_seg;
            float4 v0 = ((const float4*)src)[0];
            float4 v1 = ((const float4*)src)[1];
            float* dst = &ldsA[c & 1][st_row * LDSA_STRIDE + st_seg];
            dst[0] = v0.x; dst[1] = v0.y; dst[2] = v0.z; dst[3] = v0.w;
            dst[4] = v1.x; dst[5] = v1.y; dst[6] = v1.z; dst[7] = v1.w;
        }
#endif
        __syncthreads();

        const float m = amask[c];                      // fold mask into B
        const float* A = &ldsA[c & 1][0];

        #pragma unroll
        for (int kk = 0; kk < 16; ++kk) {
            const int kq = kk * 4 + lhalf * 2;
            v2f a[RT_PER_WAVE];
            #pragma unroll
            for (int rt = 0; rt < RT_PER_WAVE; ++rt) {
                a[rt][0] = A[(rt * 16 + l16) * LDSA_STRIDE + kq];
                a[rt][1] = A[(rt * 16 + l16) * LDSA_STRIDE + kq + 1];
            }
            #pragma unroll
            for (int t = 0; t < CT_PER_WAVE; ++t) {
                const float2 braw =
                    *(const float2*)(bb + kk * B_KK_STRIDE + t * 64);
                v2f bf;
                bf[0] = braw.x * m;
                bf[1] = braw.y * m;
                #pragma unroll
                for (int rt = 0; rt < RT_PER_WAVE; ++rt)
                    acc[rt][t] = __builtin_amdgcn_wmma_f32_16x16x4_f32(
                        false, a[rt], false, bf, (short)0, acc[rt][t],
                        false, false);
            }
        }
        bb += B_CHUNK_STRIDE;
        __syncthreads();
    }

    // ---- epilogue: + sum_m mlp_mask*mlp_cache + base, then store ----
    // mlp_cache / out are stream-once: use non-temporal to keep W_O in L2.
    const float* mmask = mlp_mask + b * NM;
    #pragma unroll
    for (int rt = 0; rt < RT_PER_WAVE; ++rt) {
        #pragma unroll
        for (int t = 0; t < CT_PER_WAVE; ++t) {
            const int col = (wave * CT_PER_WAVE + t) * 16 + l16;
            const float bval = base[b * ND + col];
            #pragma unroll
            for (int i = 0; i < 8; ++i) {
                const int row = row0 + rt * 16 + i + lhalf * 8;
                float s = acc[rt][t][i] + bval;
                const float* mc = mlp_cache + (size_t)row * NM * ND + col;
                #pragma unroll
                for (int m2 = 0; m2 < NM; ++m2)
                    s += mmask[m2] * __builtin_nontemporal_load(mc + m2 * ND);
                __builtin_nontemporal_store(s, &out[(size_t)row * ND + col]);
            }
        }
    }
}

extern "C" void kernel_launch(void* const* d_in, const int* in_sizes, int n_in,
                              void* d_out, int out_size, void* d_ws, size_t ws_size,
                              hipStream_t stream) {
    const float* mlp_cache  = (const float*)d_in[0];
    const float* attn_cache = (const float*)d_in[1];
    const float* mlp_mask   = (const float*)d_in[2];
    const float* attn_mask  = (const float*)d_in[3];
    const float* mlp_const  = (const float*)d_in[4];
    const float* attn_const = (const float*)d_in[5];
    const float* W_O        = (const float*)d_in[6];
    const float* post_bias  = (const float*)d_in[7];
    float* outp    = (float*)d_out;
    float* packedB = (float*)d_ws;                 // NK*ND floats = 28.3 MB
    float* basep   = packedB + (size_t)NK * ND;    // + B*D floats

    pack_kernel<<<(NK * ND + 255) / 256, 256, 0, stream>>>(W_O, packedB);
    base_kernel<<<(NB * ND + 255) / 256, 256, 0, stream>>>(
        mlp_mask, attn_mask, mlp_const, attn_const, post_bias, basep);
    fused_kernel<<<(NB * NP) / BLK_ROWS, THREADS, 0, stream>>>(
        mlp_cache, attn_cache, mlp_mask, attn_mask, packedB, basep, outp);
}